// MultiBoxLoss_20383914787640
// MI455X (gfx1250) — compile-verified
//
#include <hip/hip_runtime.h>
#include <hip/hip_bf16.h>
#include <float.h>

// MultiBoxLoss for MI455X (gfx1250, wave32).
// Kernel 1: one 256-thread workgroup per batch; matching, losses, and
//           hard-negative mining via an LDS bitonic sort (P padded to 2048).
// Kernel 2: one wave32 reduces per-batch partials with V_WMMA_F32_16X16X4_F32
//           (ones-matrix column-sum trick, full f32 precision, fixed order).

#define NT   256
#define MAXP 2048
#define MAXG 8

typedef __attribute__((ext_vector_type(2))) float v2f;
typedef __attribute__((ext_vector_type(8))) float v8f;

__global__ __launch_bounds__(NT) void mbox_match_loss(
    const float* __restrict__ loc_data,   // (B,P,4)
    const float* __restrict__ conf_data,  // (B,P,C)
    const float* __restrict__ priors,     // (P,4) cx,cy,w,h
    const float* __restrict__ targets,    // (B,G,5) x1,y1,x2,y2,label
    float* __restrict__ partials,         // (B,4): loss_l, loss_c, num_pos, 0
    int B, int P, int C, int G)
{
  const int b   = blockIdx.x;
  const int tid = threadIdx.x;
  if (P > MAXP) P = MAXP;
  if (G > MAXG) G = MAXG;
  int PPAD = 1; while (PPAD < P) PPAD <<= 1;   // 2046 -> 2048

  __shared__ float s_bto [MAXP];   // best_truth_overlap
  __shared__ int   s_flag[MAXP];   // best_truth_idx, later pos(1)/neg(2) flags
  __shared__ float s_ce  [MAXP];   // cross-entropy per prior
  __shared__ float s_key [MAXP];   // sort key: mine
  __shared__ int   s_idx [MAXP];   // sort payload: prior index
  __shared__ float s_tr  [MAXG][4];
  __shared__ float s_lab [MAXG];
  __shared__ float s_areat[MAXG];
  __shared__ int   s_bp  [MAXG];   // best_prior_idx per truth
  __shared__ float s_redf[NT];
  __shared__ int   s_redi[NT];

  // ---- load truths ----
  for (int g = tid; g < G; g += NT) {
    const float* t = targets + ((size_t)b * G + g) * 5;
    float x1 = t[0], y1 = t[1], x2 = t[2], y2 = t[3];
    s_tr[g][0] = x1; s_tr[g][1] = y1; s_tr[g][2] = x2; s_tr[g][3] = y2;
    s_lab[g]   = t[4];
    s_areat[g] = (x2 - x1) * (y2 - y1);
  }
  __syncthreads();

  // ---- phase A: jaccard; per-prior best truth, per-thread per-truth best prior ----
  float gbv[MAXG]; int gbi[MAXG];
  for (int g = 0; g < MAXG; ++g) { gbv[g] = -FLT_MAX; gbi[g] = 0x7fffffff; }

  for (int p = tid; p < P; p += NT) {
    float cx = priors[p*4+0], cy = priors[p*4+1];
    float w  = priors[p*4+2], h  = priors[p*4+3];
    float px1 = cx - w*0.5f, py1 = cy - h*0.5f;
    float px2 = cx + w*0.5f, py2 = cy + h*0.5f;
    float areap = (px2 - px1) * (py2 - py1);
    float bto = -FLT_MAX; int bti = 0;
    for (int g = 0; g < G; ++g) {
      float lx = fmaxf(s_tr[g][0], px1), ly = fmaxf(s_tr[g][1], py1);
      float rx = fminf(s_tr[g][2], px2), ry = fminf(s_tr[g][3], py2);
      float iw = fmaxf(rx - lx, 0.0f),  ih = fmaxf(ry - ly, 0.0f);
      float inter = iw * ih;
      float ov = inter / (s_areat[g] + areap - inter);
      if (ov > bto) { bto = ov; bti = g; }                  // first-max over g
      if (ov > gbv[g] || (ov == gbv[g] && p < gbi[g])) {    // first-max over p
        gbv[g] = ov; gbi[g] = p;
      }
    }
    s_bto[p] = bto; s_flag[p] = bti;
  }
  __syncthreads();

  // ---- phase B: reduce per-truth argmax over priors ----
  for (int g = 0; g < G; ++g) {
    s_redf[tid] = gbv[g]; s_redi[tid] = gbi[g];
    __syncthreads();
    for (int s = NT/2; s > 0; s >>= 1) {
      if (tid < s) {
        float v2 = s_redf[tid+s]; int i2 = s_redi[tid+s];
        if (v2 > s_redf[tid] || (v2 == s_redf[tid] && i2 < s_redi[tid])) {
          s_redf[tid] = v2; s_redi[tid] = i2;
        }
      }
      __syncthreads();
    }
    if (tid == 0) s_bp[g] = s_redi[0];
    __syncthreads();
  }
  // ---- phase C: force-match overrides (sequential, last-wins like scatter) ----
  if (tid == 0) {
    for (int g = 0; g < G; ++g) { int pp = s_bp[g]; s_bto[pp] = 2.0f; s_flag[pp] = g; }
  }
  __syncthreads();

  // ---- phase D: encode, smooth-L1 + GIoU term, logsumexp CE ----
  float ll = 0.0f; int np = 0;
  for (int p = tid; p < P; p += NT) {
    int   bti = s_flag[p];
    float bto = s_bto[p];
    float mx1 = s_tr[bti][0], my1 = s_tr[bti][1];
    float mx2 = s_tr[bti][2], my2 = s_tr[bti][3];
    int conf = (int)s_lab[bti] + 1;
    if (bto < 0.5f) conf = 0;
    int pos = (conf > 0);

    float cx = priors[p*4+0], cy = priors[p*4+1];
    float w  = priors[p*4+2], h  = priors[p*4+3];
    // encode(matched, prior)
    float t0 = ((mx1 + mx2) * 0.5f - cx) / (0.1f * w);
    float t1 = ((my1 + my2) * 0.5f - cy) / (0.1f * h);
    float t2 = logf((mx2 - mx1) / w) / 0.2f;
    float t3 = logf((my2 - my1) / h) / 0.2f;

    const float* L = loc_data + ((size_t)b * P + p) * 4;
    float l0 = L[0], l1 = L[1], l2 = L[2], l3 = L[3];
    float d0 = fabsf(l0-t0), d1 = fabsf(l1-t1), d2 = fabsf(l2-t2), d3 = fabsf(l3-t3);
    float sl1 = (d0 < 1.f ? 0.5f*d0*d0 : d0-0.5f) + (d1 < 1.f ? 0.5f*d1*d1 : d1-0.5f)
              + (d2 < 1.f ? 0.5f*d2*d2 : d2-0.5f) + (d3 < 1.f ? 0.5f*d3*d3 : d3-0.5f);

    // decode(loc_data, prior) -> corner box
    float dcx = cx + l0 * 0.1f * w;
    float dcy = cy + l1 * 0.1f * h;
    float dw  = w * expf(l2 * 0.2f);
    float dh  = h * expf(l3 * 0.2f);
    float bx1 = dcx - dw*0.5f, by1 = dcy - dh*0.5f;
    float bx2 = dcx + dw*0.5f, by2 = dcy + dh*0.5f;
    // giou_aligned(decoded, loc_t)  -- loc_t intentionally treated as a box, per reference
    float area1 = (bx2 - bx1) * (by2 - by1);
    float area2 = (t2 - t0) * (t3 - t1);
    float iw = fmaxf(fminf(bx2, t2) - fmaxf(bx1, t0), 0.0f);
    float ih = fmaxf(fminf(by2, t3) - fmaxf(by1, t1), 0.0f);
    float inter = iw * ih;
    float ow = fmaxf(fmaxf(bx2, t2) - fminf(bx1, t0), 0.0f);
    float oh = fmaxf(fmaxf(by2, t3) - fminf(by1, t1), 0.0f);
    float outer = ow * oh;
    float uni = area1 + area2 - inter;
    float giou = inter / uni - (outer - uni) / outer;
    if (pos) { ll += sl1 + (1.0f - giou); np += 1; }

    // cross entropy: logsumexp - target logit
    const float* cf = conf_data + ((size_t)b * P + p) * C;
    if (p + NT < P) __builtin_prefetch(cf + (size_t)NT * C, 0, 3);  // global_prefetch_b8
    float m = -FLT_MAX;
    for (int c2 = 0; c2 < C; ++c2) m = fmaxf(m, cf[c2]);
    float sum = 0.0f;
    for (int c2 = 0; c2 < C; ++c2) sum += expf(cf[c2] - m);
    float ce = m + logf(sum) - cf[conf];

    s_ce[p]   = ce;
    s_key[p]  = pos ? 0.0f : ce;   // mine
    s_idx[p]  = p;
    s_flag[p] = pos;               // reuse as flags: bit0 = pos
  }
  for (int p = P + tid; p < PPAD; p += NT) { s_key[p] = -FLT_MAX; s_idx[p] = p; }

  // ---- reduce num_pos and loss_l (fixed-order trees -> deterministic) ----
  s_redf[tid] = ll; s_redi[tid] = np;
  __syncthreads();
  for (int s = NT/2; s > 0; s >>= 1) {
    if (tid < s) { s_redf[tid] += s_redf[tid+s]; s_redi[tid] += s_redi[tid+s]; }
    __syncthreads();
  }
  float ll_total = s_redf[0];
  int   np_total = s_redi[0];
  int num_neg = 3 * np_total; if (num_neg > P - 1) num_neg = P - 1;

  // ---- bitonic sort of (mine, idx): mine desc, idx asc => stable argsort rank ----
  for (int k = 2; k <= PPAD; k <<= 1) {
    for (int j = k >> 1; j > 0; j >>= 1) {
      for (int i = tid; i < PPAD; i += NT) {
        int ixj = i ^ j;
        if (ixj > i) {
          float ka = s_key[i];   int ia = s_idx[i];
          float kb = s_key[ixj]; int ib = s_idx[ixj];
          bool before = (kb > ka) || (kb == ka && ib < ia);  // s[ixj] ranks before s[i]
          bool up = ((i & k) == 0);
          if (before == up) {
            s_key[i] = kb; s_idx[i] = ib;
            s_key[ixj] = ka; s_idx[ixj] = ia;
          }
        }
      }
      __syncthreads();
    }
  }

  // ---- mark negatives (rank < num_neg), then loss_c = sum ce over pos|neg ----
  for (int r = tid; r < num_neg; r += NT) {
    int idx = s_idx[r];
    if (idx < P) s_flag[idx] |= 2;   // each idx unique -> race-free
  }
  __syncthreads();
  float lc = 0.0f;
  for (int p = tid; p < P; p += NT) if (s_flag[p]) lc += s_ce[p];
  s_redf[tid] = lc;
  __syncthreads();
  for (int s = NT/2; s > 0; s >>= 1) {
    if (tid < s) s_redf[tid] += s_redf[tid+s];
    __syncthreads();
  }
  if (tid == 0) {
    partials[(size_t)b*4 + 0] = ll_total;
    partials[(size_t)b*4 + 1] = s_redf[0];
    partials[(size_t)b*4 + 2] = (float)np_total;
    partials[(size_t)b*4 + 3] = 0.0f;
  }
}

// One wave32. Column sums of partials(B x 4) via C += ones(16x4) x Bchunk(4x16)
// using V_WMMA_F32_16X16X4_F32. A is all-ones so C[0][j] = sum_k B[k][j];
// only the documented "N = lane, M=0 in C VGPR0" mapping matters.
__global__ __launch_bounds__(32) void mbox_finalize(
    const float* __restrict__ partials, float* __restrict__ out, int B)
{
  const int lane = threadIdx.x;          // 0..31, EXEC all ones
  const int col  = lane & 15;            // N
  const int kb   = (lane < 16) ? 0 : 2;  // K rows held by this half-wave
  v2f a; a[0] = 1.0f; a[1] = 1.0f;       // A = ones(16x4), layout-independent
  v8f c = {};
  const int chunks = (B + 3) >> 2;
  for (int t = 0; t < chunks; ++t) {
    int r0 = t*4 + kb, r1 = r0 + 1;
    v2f bv;
    bv[0] = (col < 3 && r0 < B) ? partials[(size_t)r0*4 + col] : 0.0f;
    bv[1] = (col < 3 && r1 < B) ? partials[(size_t)r1*4 + col] : 0.0f;
    c = __builtin_amdgcn_wmma_f32_16x16x4_f32(
        /*neg_a=*/false, a, /*neg_b=*/false, bv,
        /*c_mod=*/(short)0, c, /*reuse_a=*/false, /*reuse_b=*/false);
  }
  __shared__ float s[16];
  if (lane < 16) s[lane] = c[0];         // C row M=0: lane = N
  __syncthreads();
  if (lane == 0) {
    float ll = s[0], lc = s[1], n = s[2];
    out[0] = ll / n;
    out[1] = lc / n;
  }
}

extern "C" void kernel_launch(void* const* d_in, const int* in_sizes, int n_in,
                              void* d_out, int out_size, void* d_ws, size_t ws_size,
                              hipStream_t stream) {
  (void)n_in; (void)out_size; (void)ws_size;
  const float* loc  = (const float*)d_in[0];  // (B,P,4)
  const float* conf = (const float*)d_in[1];  // (B,P,C)
  const float* pri  = (const float*)d_in[2];  // (P,4)
  const float* tgt  = (const float*)d_in[3];  // (B,G,5)
  // d_in[4] = conflict_flag, unused by the reference math

  int B = in_sizes[4];                 // conflict_flag has shape (B,)
  int P = in_sizes[2] / 4;
  int C = in_sizes[1] / (B * P);
  int G = in_sizes[3] / (B * 5);

  float* partials = (float*)d_ws;      // B*4 floats, fully written each call

  mbox_match_loss<<<B, NT, 0, stream>>>(loc, conf, pri, tgt, partials, B, P, C, G);
  mbox_finalize<<<1, 32, 0, stream>>>(partials, (float*)d_out, B);
}